// RoILocalDFVSPool3d_6657199309414
// MI455X (gfx1250) — compile-verified
//
#include <hip/hip_runtime.h>
#include <stdint.h>

typedef __attribute__((ext_vector_type(2))) float v2f;
typedef __attribute__((ext_vector_type(8))) float v8f;

#define NPTS   131072
#define NBOX   256           // B*M
#define NSEG   256           // point segments
#define SEGPTS (NPTS / NSEG) // 512 points per segment
#define ITERS  (SEGPTS / 16) // 32 WMMA chunks of 16 points
#define NOUT   512
#define NTILE  16            // 16 boxes per tile, 16 tiles

__device__ __forceinline__ unsigned ballot32(bool p) {
#if __has_builtin(__builtin_amdgcn_ballot_w32)
  return __builtin_amdgcn_ballot_w32(p);
#else
  return (unsigned)__ballot((int)p);
#endif
}

// ---------------------------------------------------------------------------
// Async stage: lanes 0..15 each pull one float4 point row into LDS.
// ---------------------------------------------------------------------------
__device__ __forceinline__ void issue_async16(const float* __restrict__ points,
                                              int p0, unsigned lds_byte) {
  const int lane = threadIdx.x;
  if (lane < 16) {
    const float* gp = points + (size_t)(p0 + lane) * 4;
    unsigned laddr = lds_byte + (unsigned)lane * 16u;
    asm volatile("global_load_async_to_lds_b128 %0, %1, off"
                 :: "v"(laddr), "v"(gp)
                 : "memory");
  }
}

// ---------------------------------------------------------------------------
// Main pass: WRITE=false -> per-(box,segment) counts; WRITE=true -> emit idx.
// One wave (32 threads) per block; block = (tile of 16 boxes) x (segment).
// ---------------------------------------------------------------------------
template<bool WRITE>
__global__ void pool_pass(const float* __restrict__ points,
                          const float* __restrict__ boxes,
                          unsigned* __restrict__ counts,
                          const unsigned* __restrict__ offs,
                          int* __restrict__ out_idx) {
  __shared__ float4 pbuf[2][16];

  const int lane   = threadIdx.x;           // 0..31
  const bool lol   = lane < 16;
  const int tile   = blockIdx.x;            // 0..15
  const int tile16 = tile * 16;
  const int seg    = blockIdx.y;            // 0..255
  const int p0base = seg * SEGPTS;

  // ---- per-box setup: lane m holds box (tile16 + (m&15)) --------------------
  const int m = lane & 15;
  const float* bx = boxes + (size_t)(tile16 + m) * 7;
  const float cx = bx[0], cy = bx[1], cz = bx[2];
  const float d0 = bx[3], d1 = bx[4], d2 = bx[5];
  const float hd = -(bx[6] + 1.57079632679489662f);   // pcdet heading
  const float c  = __cosf(hd);
  const float s  = __sinf(hd);
  // pcdet dims = (d1, d0, d2), +2*POOL_EXTRA_WIDTH, halved
  const float hx = 0.5f * (d1 + 2.0f);
  const float hy = 0.5f * (d0 + 2.0f);
  const float hz = 0.5f * (d2 + 2.0f);
  const float tx = -(cx * c + cy * s);
  const float ty =  (cx * s - cy * c);

  // A matrices (16x4 f32): lanes 0-15 hold K=0(v0),K=1(v1); lanes 16-31 K=2,K=3
  v2f alx = { lol ?  c    : 0.0f, lol ? s    : tx  };
  v2f aly = { lol ? (-s)  : 0.0f, lol ? c    : ty  };
  v2f adz = { lol ?  0.0f : 1.0f, lol ? 0.0f : -cz };

  // Half-dims for the 8 boxes this lane's D rows cover (r / r+8 per lane half)
  float hx8[8], hy8[8], hz8[8];
#pragma unroll
  for (int r = 0; r < 8; ++r) {
    int src = lol ? r : (r + 8);
    hx8[r] = __shfl(hx, src, 32);
    hy8[r] = __shfl(hy, src, 32);
    hz8[r] = __shfl(hz, src, 32);
  }
  const float batchf = (float)(tile >> 3);  // 8 tiles per batch
  const unsigned ltm = (1u << (lane & 15)) - 1u;

  unsigned acc[16];
  if (WRITE) {
#pragma unroll
    for (int j = 0; j < 16; ++j)
      acc[j] = offs[(size_t)(tile16 + j) * NSEG + seg];
  } else {
#pragma unroll
    for (int j = 0; j < 16; ++j) acc[j] = 0u;
  }

  const unsigned ldsbase = (unsigned)(uintptr_t)(void*)pbuf;
  issue_async16(points, p0base, ldsbase);   // prologue: chunk 0

  const v8f czero = {0.f,0.f,0.f,0.f,0.f,0.f,0.f,0.f};

  for (int it = 0; it < ITERS; ++it) {
    if (it + 1 < ITERS) {
      issue_async16(points, p0base + (it + 1) * 16,
                    ldsbase + (unsigned)((it + 1) & 1) * 256u);
      asm volatile("s_wait_asynccnt 0x1" ::: "memory");   // chunk 'it' landed
    } else {
      asm volatile("s_wait_asynccnt 0x0" ::: "memory");
    }

    const int   pidx = p0base + it * 16 + (lane & 15);
    const float4 pt  = pbuf[it & 1][lane & 15];

    // B matrix (4x16): lanes 0-15: v0=px,v1=py ; lanes 16-31: v0=pz,v1=1
    v2f bm = { lol ? pt.y : pt.w, lol ? pt.z : 1.0f };

    v8f lxv = __builtin_amdgcn_wmma_f32_16x16x4_f32(false, alx, false, bm,
                                                    (short)0, czero, false, false);
    v8f lyv = __builtin_amdgcn_wmma_f32_16x16x4_f32(false, aly, false, bm,
                                                    (short)0, czero, false, false);
    v8f dzv = __builtin_amdgcn_wmma_f32_16x16x4_f32(false, adz, false, bm,
                                                    (short)0, czero, false, false);

    const bool bok = (pt.x == batchf);

#pragma unroll
    for (int r = 0; r < 8; ++r) {
      bool ins = (__builtin_fabsf(lxv[r]) <= hx8[r]) &&
                 (__builtin_fabsf(lyv[r]) <= hy8[r]) &&
                 (__builtin_fabsf(dzv[r]) <= hz8[r]) && bok;
      unsigned mask   = ballot32(ins);
      unsigned lomask = mask & 0xFFFFu;
      unsigned himask = mask >> 16;
      if (WRITE) {
        unsigned posLo = acc[r]     + (unsigned)__popc(lomask & ltm);
        unsigned posHi = acc[r + 8] + (unsigned)__popc(himask & ltm);
        unsigned pos   = lol ? posLo : posHi;
        int boxg = tile16 + (lol ? r : r + 8);
        if (ins && pos < NOUT)
          out_idx[(size_t)boxg * NOUT + pos] = pidx;
      }
      acc[r]     += (unsigned)__popc(lomask);
      acc[r + 8] += (unsigned)__popc(himask);
    }
  }

  if (!WRITE && lane == 0) {
#pragma unroll
    for (int j = 0; j < 16; ++j)
      counts[(size_t)(tile16 + j) * NSEG + seg] = acc[j];
  }
}

// ---------------------------------------------------------------------------
// Scan over segments per box -> exclusive offsets + pooled_num.
// ---------------------------------------------------------------------------
__global__ void scan_pass(const unsigned* __restrict__ counts,
                          unsigned* __restrict__ offs,
                          int* __restrict__ pooled_num) {
  const int b = threadIdx.x;   // 256 threads, 1 block
  unsigned run = 0;
  for (int s = 0; s < NSEG; ++s) {
    unsigned cnt = counts[(size_t)b * NSEG + s];
    offs[(size_t)b * NSEG + s] = run;
    run += cnt;
  }
  pooled_num[b] = (int)(run < (unsigned)NOUT ? run : (unsigned)NOUT);
}

__global__ void fill_zero(int* __restrict__ p, int n) {
  int i = blockIdx.x * blockDim.x + threadIdx.x;
  if (i < n) p[i] = 0;
}

// ---------------------------------------------------------------------------
extern "C" void kernel_launch(void* const* d_in, const int* in_sizes, int n_in,
                              void* d_out, int out_size, void* d_ws, size_t ws_size,
                              hipStream_t stream) {
  const float* points = (const float*)d_in[0];   // (N,4): [bidx,x,y,z]
  const float* boxes  = (const float*)d_in[1];   // (B,M,7)
  int* out_idx = (int*)d_out;                    // [256][512]
  int* out_num = out_idx + (size_t)NBOX * NOUT;  // [256]

  unsigned* counts = (unsigned*)d_ws;            // [256][256]
  unsigned* offs   = counts + (size_t)NBOX * NSEG;

  fill_zero<<<(NBOX * NOUT + 255) / 256, 256, 0, stream>>>(out_idx, NBOX * NOUT);

  dim3 grid(NTILE, NSEG);
  pool_pass<false><<<grid, 32, 0, stream>>>(points, boxes, counts, nullptr, nullptr);
  scan_pass<<<1, NBOX, 0, stream>>>(counts, offs, out_num);
  pool_pass<true><<<grid, 32, 0, stream>>>(points, boxes, nullptr, offs, out_idx);
}